// Dropout_partial_binary_3650722201830
// MI455X (gfx1250) — compile-verified
//
#include <hip/hip_runtime.h>

typedef __attribute__((ext_vector_type(2))) float v2f;
typedef __attribute__((ext_vector_type(8))) float v8f;

#define K_CH 128   // gathered channels (matches reference)
#define WAVE 32    // gfx1250 is wave32

// One wave owns 16 rows. Phase 1: accumulate dropped-sum of 16 rows via
// V_WMMA_F32_16X16X4_F32 with an all-ones B matrix (D[m,*] = sum_k A[m,k]).
// Phase 2: broadcast comp[m] = dropped_sum[m]/K via lane shuffles.
// Phase 3: stream-copy each 1KB row X->out with float4, then overwrite the
// K gathered slots with (masked?0:v)+comp. Same-wave stores are ordered, so
// the scatter lands after the bulk copy.
__global__ __launch_bounds__(256)
void dropout_partial_binary_kernel(const float* __restrict__ X,
                                   const int* __restrict__ idx,
                                   const unsigned char* __restrict__ mask,
                                   float* __restrict__ out,
                                   int R, int F, int K)
{
    __shared__ int sIdx[K_CH];
    const int tid = threadIdx.x;
    if (tid < K) sIdx[tid] = idx[tid];
    __syncthreads();

    const int lane   = tid & 31;
    const int waveId = tid >> 5;
    const long rowBase = ((long)blockIdx.x * 8 + waveId) * 16;
    if (rowBase + 15 >= R) return;   // uniform per wave: EXEC stays all-ones

    // ---- Phase 1: WMMA reduction of dropped mass over K, 16 rows at once ----
    // A-matrix 16x4 f32 layout: lanes 0-15 -> M=lane, K={4c+0,4c+1};
    //                           lanes 16-31 -> M=lane-16, K={4c+2,4c+3}.
    const int m  = lane & 15;
    const int hi = lane >> 4;
    const float*         xRowM = X    + (rowBase + m) * (long)F;
    const unsigned char* mRowM = mask + (rowBase + m) * (long)K;

    v8f acc = {};
    v2f bOnes; bOnes[0] = 1.0f; bOnes[1] = 1.0f;

    #pragma unroll 4
    for (int c = 0; c < K_CH / 4; ++c) {
        const int k0 = 4 * c + 2 * hi;
        const int c0 = sIdx[k0];
        const int c1 = sIdx[k0 + 1];
        const float v0 = xRowM[c0];
        const float v1 = xRowM[c1];
        v2f a;
        a[0] = mRowM[k0]     ? v0 : 0.0f;   // dropped contributions only
        a[1] = mRowM[k0 + 1] ? v1 : 0.0f;
        // D = A x ones + C  ->  per-row dropped-sum accumulation (f32 exact)
        acc = __builtin_amdgcn_wmma_f32_16x16x4_f32(
                  false, a, false, bOnes, (short)0, acc, false, false);
    }

    // ---- Phase 2: broadcast comp[r] to all lanes ----
    // C/D 16x16 f32 layout: VGPR v, lanes 0-15 -> M=v, lanes 16-31 -> M=v+8.
    float comp[16];
    const float invK = 1.0f / (float)K;
    #pragma unroll
    for (int r = 0; r < 16; ++r)
        comp[r] = __shfl(acc[r & 7], (r >> 3) * 16, 32) * invK;

    // ---- Phase 3: stream copy + scatter update ----
    for (int r = 0; r < 16; ++r) {
        const long row = rowBase + r;
        const float*         xRow = X    + row * (long)F;
        const unsigned char* mRow = mask + row * (long)K;
        float*               oRow = out  + row * (long)F;

        // bulk copy: each lane moves float4s (fully coalesced 512B/instr)
        for (int f = lane * 4; f < F; f += WAVE * 4) {
            float4 vv = *(const float4*)(xRow + f);
            *(float4*)(oRow + f) = vv;
        }
        // gathered overwrite: consecutive lanes -> consecutive idx entries
        #pragma unroll
        for (int jj = 0; jj < K_CH / WAVE; ++jj) {
            const int j   = jj * WAVE + lane;
            const int col = sIdx[j];
            const float v = xRow[col];                 // hits WGP$/L2
            oRow[col] = (mRow[j] ? 0.0f : v) + comp[r];
        }
    }
}

extern "C" void kernel_launch(void* const* d_in, const int* in_sizes, int n_in,
                              void* d_out, int out_size, void* d_ws, size_t ws_size,
                              hipStream_t stream) {
    const float*         X    = (const float*)d_in[0];
    const int*           idx  = (const int*)d_in[1];
    const unsigned char* mask = (const unsigned char*)d_in[2];   // jnp bool -> 1 byte
    float*               out  = (float*)d_out;

    const int  K = in_sizes[1];                 // 128
    const long R = (long)in_sizes[2] / K;       // B*C*T = 262144 rows
    const int  F = (int)((long)in_sizes[0] / R);// 256

    // 256 threads = 8 waves; 16 rows per wave -> 128 rows per block
    const long rowsPerBlock = 128;
    const int  blocks = (int)((R + rowsPerBlock - 1) / rowsPerBlock);
    dropout_partial_binary_kernel<<<blocks, 256, 0, stream>>>(
        X, idx, mask, out, (int)R, F, K);
}